// lower_module_79491254714775
// MI455X (gfx1250) — compile-verified
//
#include <hip/hip_runtime.h>
#include <math.h>

// ---------------- problem constants ----------------
constexpr int FXY     = 7;
constexpr int CCH     = 512;
constexpr int HH      = 64;
constexpr int WW      = 64;
constexpr int POOLLEN = CCH * FXY * FXY;   // 25088
constexpr int NSEG    = 6;
constexpr int K_TOTAL = POOLLEN * NSEG;    // 150528
constexpr int KCHUNKS = K_TOTAL / 4;       // 37632 (POOLLEN % 4 == 0 -> chunks never straddle segments)
constexpr int NOUT    = 1024;

typedef __attribute__((ext_vector_type(2))) float v2f;
typedef __attribute__((ext_vector_type(8))) float v8f;

// ---------------- 1) box/region computation ----------------
// regions[r] = {y0, y1, x0, x1};  r: 0=scene, 1..3=human[i], 4+4*i+j=union(person i, obj j)
__global__ void boxes_kernel(const float* __restrict__ pbb, const float* __restrict__ obb,
                             const float* __restrict__ ratio, int* __restrict__ regions) {
  if (threadIdx.x != 0 || blockIdx.x != 0) return;
  const float rx = ratio[0], ry = ratio[1];
  int pb[3][4], ob[4][4];
  for (int i = 0; i < 3; ++i) {
    pb[i][0] = (int)rintf(pbb[i * 4 + 0] * rx);  // x1
    pb[i][1] = (int)rintf(pbb[i * 4 + 1] * ry);  // y1
    pb[i][2] = (int)rintf(pbb[i * 4 + 2] * rx);  // x2
    pb[i][3] = (int)rintf(pbb[i * 4 + 3] * ry);  // y2
  }
  for (int j = 0; j < 4; ++j) {
    ob[j][0] = (int)rintf(obb[j * 4 + 0] * rx);
    ob[j][1] = (int)rintf(obb[j * 4 + 1] * ry);
    ob[j][2] = (int)rintf(obb[j * 4 + 2] * rx);
    ob[j][3] = (int)rintf(obb[j * 4 + 3] * ry);
  }
  regions[0] = 0; regions[1] = HH; regions[2] = 0; regions[3] = WW;   // scene
  for (int i = 0; i < 3; ++i) {
    int* r = regions + (1 + i) * 4;
    r[0] = pb[i][1]; r[1] = pb[i][3]; r[2] = pb[i][0]; r[3] = pb[i][2];
  }
  for (int i = 0; i < 3; ++i)
    for (int j = 0; j < 4; ++j) {
      int* r = regions + (4 + 4 * i + j) * 4;
      r[0] = min(pb[i][1], ob[j][1]); r[1] = max(pb[i][3], ob[j][3]);
      r[2] = min(pb[i][0], ob[j][0]); r[3] = max(pb[i][2], ob[j][2]);
    }
}

// ---------------- 2) adaptive max pool (PyTorch bins) ----------------
__global__ void pool_kernel(const float* __restrict__ feat, const int* __restrict__ regions,
                            float* __restrict__ pool) {
  int tid = blockIdx.x * blockDim.x + threadIdx.x;
  if (tid >= 16 * CCH * FXY * FXY) return;
  int reg = tid / (CCH * 49);
  int rem = tid % (CCH * 49);
  int c   = rem / 49;
  int ij  = rem % 49;
  int oi  = ij / 7, oj = ij % 7;
  const int* r = regions + reg * 4;
  int y0 = r[0], y1 = r[1], x0 = r[2], x1 = r[3];
  int h = y1 - y0, w = x1 - x0;
  int rs = y0 + (oi * h) / 7;
  int re = y0 + ((oi + 1) * h + 6) / 7;   // ceil
  int cs = x0 + (oj * w) / 7;
  int ce = x0 + ((oj + 1) * w + 6) / 7;   // ceil
  float m = -__builtin_inff();
  const float* fc = feat + (size_t)c * (HH * WW);
  for (int y = rs; y < re; ++y)
    for (int x = cs; x < ce; ++x)
      m = fmaxf(m, fc[y * WW + x]);
  pool[(size_t)reg * POOLLEN + c * 49 + ij] = m;
}

// segment map: row i of A = concat(human[i], obj[4i..4i+3], scene)
__device__ __forceinline__ float aval(const float* __restrict__ pool, int i, int k) {
  int s = k / POOLLEN, r = k % POOLLEN;
  int reg = (s == 0) ? (1 + i) : ((s <= 4) ? (4 + 4 * i + (s - 1)) : 0);
  return pool[(size_t)reg * POOLLEN + r];
}

// ---------------- 3) pre-pack WMMA A-fragments ----------------
// ISA 7.12.2, 32-bit A 16x4: VGPR0: lanes0-15 K=0 / lanes16-31 K=2; VGPR1: K=1 / K=3; M = lane%16.
// afrag[kc*32 + lane] = float2{VGPR0, VGPR1}. Rows M>=3 are zero (persons padded to M=16).
__global__ void afrag_kernel(const float* __restrict__ pool, float2* __restrict__ afrag) {
  int tid = blockIdx.x * blockDim.x + threadIdx.x;
  if (tid >= KCHUNKS * 32) return;
  int lane = tid & 31;
  int kc   = tid >> 5;
  int M    = lane & 15;
  int klo  = (lane < 16) ? 0 : 2;
  float v0 = 0.f, v1 = 0.f;
  if (M < 3) {
    int k0 = kc * 4 + klo;
    v0 = aval(pool, M, k0);
    v1 = aval(pool, M, k0 + 1);
  }
  float2 rv; rv.x = v0; rv.y = v1;
  afrag[tid] = rv;
}

// ---------------- 4) bias init ----------------
__global__ void init_out_kernel(const float* __restrict__ b, float* __restrict__ out) {
  int t = blockIdx.x * blockDim.x + threadIdx.x;
  if (t < 3 * NOUT) out[t] = b[t % NOUT];
}

// ---------------- 5) split-K WMMA GEMM: out[3,1024] += A[16,K] x W[K,1024] ----------------
// grid.x = 64 N-tiles of 16 cols, grid.y = K split; 4 waves/block.
// KCHUNKS = 37632 = (KSPLIT*WAVES_PER_BLOCK) * 294 exactly -> compile-time trip count,
// pure pointer-increment addressing, uniform (SGPR) loop control.
constexpr int WAVES_PER_BLOCK = 4;
constexpr int KSPLIT          = 32;
constexpr int KSTRIDE         = KSPLIT * WAVES_PER_BLOCK;   // 128 chunks
constexpr int KITERS          = KCHUNKS / KSTRIDE;          // 294 (exact)
static_assert(KCHUNKS % KSTRIDE == 0, "exact split-K");

__global__ void __launch_bounds__(WAVES_PER_BLOCK * 32)
gemm_kernel(const float* __restrict__ Wfc, const float2* __restrict__ afrag,
            float* __restrict__ out) {
  const int lane = threadIdx.x & 31;
  // Uniform per wave -> force into SGPR so the loop is scalar-controlled (no exec churn).
  const int kgrp = __builtin_amdgcn_readfirstlane(blockIdx.y * WAVES_PER_BLOCK +
                                                  (threadIdx.x >> 5));
  const int n    = blockIdx.x * 16 + (lane & 15);
  const int krlo = (lane < 16) ? 0 : 2;   // B 4x16: VGPR0 rows K=0/2, VGPR1 rows K=1/3

  // Streaming pointers with constant strides (strength-reduced addressing).
  const float2* ap = afrag + (size_t)kgrp * 32 + lane;
  const float*  bp = Wfc + ((size_t)kgrp * 4 + krlo) * NOUT + n;
  constexpr size_t ASTEP = (size_t)KSTRIDE * 32;          // float2 elements
  constexpr size_t BSTEP = (size_t)KSTRIDE * 4 * NOUT;    // float elements (2 MB jump)

  v8f c = {0.f, 0.f, 0.f, 0.f, 0.f, 0.f, 0.f, 0.f};

#pragma unroll 2
  for (int it = 0; it < KITERS; ++it) {
    float2 a2 = ap[0];                    // coalesced b64, L2-resident A stream
    v2f a; a.x = a2.x; a.y = a2.y;
    v2f b; b.x = bp[0]; b.y = bp[NOUT];   // two W_fc rows, half-wave 64B contiguity
    // Unconditional speculative prefetch of next iteration's W line
    // (translation failures are silently dropped; no exec-mask churn).
    __builtin_prefetch(bp + BSTEP, 0, 0);
    c = __builtin_amdgcn_wmma_f32_16x16x4_f32(
        /*neg_a=*/false, a, /*neg_b=*/false, b,
        /*c_mod=*/(short)0, c, /*reuse_a=*/false, /*reuse_b=*/false);
    ap += ASTEP;
    bp += BSTEP;
  }

  // C/D layout: VGPR v, lanes 0-15 -> M=v. Rows 0..2 are the 3 persons.
  if (lane < 16) {
    atomicAdd(&out[0 * NOUT + n], c[0]);
    atomicAdd(&out[1 * NOUT + n], c[1]);
    atomicAdd(&out[2 * NOUT + n], c[2]);
  }
}

// ---------------- launch ----------------
extern "C" void kernel_launch(void* const* d_in, const int* in_sizes, int n_in,
                              void* d_out, int out_size, void* d_ws, size_t ws_size,
                              hipStream_t stream) {
  const float* feature = (const float*)d_in[0];   // [1,512,64,64]
  const float* pbb     = (const float*)d_in[1];   // [3,4]
  const float* obb     = (const float*)d_in[2];   // [4,4]
  const float* ratio   = (const float*)d_in[3];   // [2]
  const float* Wfc     = (const float*)d_in[4];   // [150528,1024]
  const float* bfc     = (const float*)d_in[5];   // [1024]
  float* out = (float*)d_out;                     // [3,1024]

  char* ws = (char*)d_ws;
  int*    regions = (int*)ws;                                        // 16*4 ints
  float*  pool    = (float*)(ws + 256);                              // 16*25088 f32 (1.6 MB)
  float2* afrag   = (float2*)(ws + 256 + (size_t)16 * POOLLEN * 4);  // 37632*32 float2 (9.6 MB)

  boxes_kernel<<<1, 1, 0, stream>>>(pbb, obb, ratio, regions);

  int poolThreads = 16 * CCH * FXY * FXY;
  pool_kernel<<<(poolThreads + 255) / 256, 256, 0, stream>>>(feature, regions, pool);

  int fragThreads = KCHUNKS * 32;
  afrag_kernel<<<(fragThreads + 255) / 256, 256, 0, stream>>>(pool, afrag);

  init_out_kernel<<<(3 * NOUT + 255) / 256, 256, 0, stream>>>(bfc, out);

  dim3 grid(NOUT / 16, KSPLIT);
  gemm_kernel<<<grid, WAVES_PER_BLOCK * 32, 0, stream>>>(Wfc, afrag, out);
}